// TopKMoEGate_53154515256360
// MI455X (gfx1250) — compile-verified
//
#include <hip/hip_runtime.h>
#include <math.h>

typedef __attribute__((ext_vector_type(2))) float v2f;
typedef __attribute__((ext_vector_type(8))) float v8f;

#define WAVES   8
#define TPW     16            // tokens per wave (WMMA M tile)
#define TOK_WG  (WAVES * TPW) // 128 tokens per workgroup
#define KSTAGE  32            // K chunk staged through LDS (double buffered)
#define NT      4             // 4 N-tiles of 16 -> E = 64 experts
#define LSTR    36            // padded stage row stride (floats); 36*4B, conflict-free
#define ESTR    68            // padded epilogue row stride (floats)

// ---- CDNA5 async memory->LDS copy (ASYNCcnt path), inline asm ----
__device__ __forceinline__ void async_ld_b128(const float* gaddr, const float* lds_ptr) {
    const unsigned lds_off = (unsigned)(size_t)lds_ptr;   // low 32 bits = LDS offset
    asm volatile("global_load_async_to_lds_b128 %0, %1, off"
                 :: "v"(lds_off), "v"(gaddr) : "memory");
}
__device__ __forceinline__ void wait_async_le6() {
    asm volatile("s_wait_asynccnt 0x6" ::: "memory");
}
__device__ __forceinline__ void wait_async_0() {
    asm volatile("s_wait_asynccnt 0x0" ::: "memory");
}

__global__ __launch_bounds__(256)
void moe_gate_topk_kernel(const float* __restrict__ x,
                          const float* __restrict__ gw,
                          const float* __restrict__ nwt,
                          const float* __restrict__ noise,
                          float* __restrict__ out_w,
                          int*   __restrict__ out_idx,
                          int D)
{
    __shared__ float xs[2 * TOK_WG * LSTR]; // x stages; reused as logits scratch (128 x ESTR fits)
    __shared__ float ws[2 * 64 * LSTR];     // gate_w stages
    __shared__ float nw_s[64];
    __shared__ float tw1[TOK_WG], tw2[TOK_WG];
    __shared__ int   ti1[TOK_WG], ti2[TOK_WG];

    const int tid     = threadIdx.x;
    const int lane    = tid & 31;   // wave32
    const int wave    = tid >> 5;
    const int tokBase = blockIdx.x * TOK_WG;

    if (tid < 64) nw_s[tid] = nwt[tid];

    v8f acc[NT];
    #pragma unroll
    for (int n = 0; n < NT; ++n) acc[n] = (v8f){0.f,0.f,0.f,0.f,0.f,0.f,0.f,0.f};

    const int nstage = D / KSTAGE;

    // Per-stage async staging: 6 x b128 ops per thread (4 for x block, 2 for gate_w block)
    auto issue_stage = [&](int s, int b) {
        const int kBase = s * KSTAGE;
        float* xb = &xs[b * TOK_WG * LSTR];
        float* wb = &ws[b * 64 * LSTR];
        #pragma unroll
        for (int j = 0; j < 4; ++j) {              // x: 128 rows x 32 cols
            const int q  = tid + j * 256;
            const int r  = q >> 3;
            const int c4 = (q & 7) << 2;
            async_ld_b128(x + (size_t)(tokBase + r) * D + kBase + c4, &xb[r * LSTR + c4]);
        }
        #pragma unroll
        for (int j = 0; j < 2; ++j) {              // gate_w: 64 rows x 32 cols (L2-resident)
            const int q  = tid + j * 256;
            const int r  = q >> 3;
            const int c4 = (q & 7) << 2;
            async_ld_b128(gw + (size_t)r * D + kBase + c4, &wb[r * LSTR + c4]);
        }
    };

    issue_stage(0, 0);                              // prologue: stage 0 in flight

    for (int s = 0; s < nstage; ++s) {
        const int b = s & 1;
        if (s + 1 < nstage) {
            issue_stage(s + 1, b ^ 1);              // keep next stage's HBM traffic in flight
            wait_async_le6();                       // stage s landed; stage s+1 still flying
        } else {
            wait_async_0();
        }
        __syncthreads();

        // WMMA f32 16x16x4: A lanes0-15 K={0,1}, lanes16-31 K={2,3}; B mirrors with N=lane&15
        const float* xb = &xs[b * TOK_WG * LSTR];
        const float* wb = &ws[b * 64 * LSTR];
        const int arow  = wave * TPW + (lane & 15);
        const int khalf = (lane >> 4) << 1;
        #pragma unroll
        for (int kk = 0; kk < KSTAGE; kk += 4) {
            const v2f a = *(const v2f*)(&xb[arow * LSTR + kk + khalf]);
            #pragma unroll
            for (int n = 0; n < NT; ++n) {
                const int ecol = n * 16 + (lane & 15);
                const v2f bfrag = *(const v2f*)(&wb[ecol * LSTR + kk + khalf]);
                acc[n] = __builtin_amdgcn_wmma_f32_16x16x4_f32(
                    false, a, false, bfrag, (short)0, acc[n], false, false);
            }
        }
        __syncthreads();   // all waves done with buffer b before stage s+2 overwrites it
    }

    // Spill logits to LDS (reuse xs). C/D layout: VGPR v -> M = v (lanes 0-15) / v+8 (16-31)
    float* lg = xs;
    #pragma unroll
    for (int n = 0; n < NT; ++n) {
        const int e = n * 16 + (lane & 15);
        #pragma unroll
        for (int v = 0; v < 8; ++v) {
            const int m = v + ((lane >> 4) << 3);
            lg[(wave * TPW + m) * ESTR + e] = acc[n][v];
        }
    }
    __syncthreads();

    // Per-token noisy top-2 + exact 2-way softmax (lanes 0..15 of each wave, 1 token each)
    if (lane < 16) {
        const int    t   = wave * TPW + lane;
        const size_t tok = (size_t)tokBase + t;
        float v1 = -INFINITY, v2v = -INFINITY;
        int   i1 = 0, i2 = 0;
        for (int e = 0; e < 64; ++e) {
            const float lv = lg[t * ESTR + e] + noise[tok * 64 + e] * nw_s[e];
            if (lv > v1)       { v2v = v1; i2 = i1; v1 = lv; i1 = e; }
            else if (lv > v2v) { v2v = lv; i2 = e; }
        }
        const float d   = expf(v2v - v1);
        const float inv = 1.0f / (1.0f + d);
        tw1[t] = inv;  tw2[t] = d * inv;
        ti1[t] = i1;   ti2[t] = i2;
    }
    __syncthreads();

    // Coalesced dense write of weights [128 x 64] (zero-fills non-top entries)
    #pragma unroll
    for (int j = 0; j < 32; ++j) {
        const int f = tid + j * 256;
        const int t = f >> 6;
        const int e = f & 63;
        float v = 0.0f;
        if (e == ti1[t])      v = tw1[t];
        else if (e == ti2[t]) v = tw2[t];
        out_w[((size_t)tokBase + t) * 64 + e] = v;
    }
    if (tid < TOK_WG) {
        out_idx[((size_t)tokBase + tid) * 2 + 0] = ti1[tid];
        out_idx[((size_t)tokBase + tid) * 2 + 1] = ti2[tid];
    }
}

extern "C" void kernel_launch(void* const* d_in, const int* in_sizes, int n_in,
                              void* d_out, int out_size, void* d_ws, size_t ws_size,
                              hipStream_t stream) {
    const float* x     = (const float*)d_in[0];
    const float* gw    = (const float*)d_in[1];
    const float* nwt   = (const float*)d_in[2];
    const float* noise = (const float*)d_in[3];

    const int    E    = in_sizes[2];                       // 64
    const size_t Ntok = (size_t)in_sizes[3] / E;           // B*S = 16384
    const int    D    = (int)((size_t)in_sizes[0] / Ntok); // 1024

    float* out_w   = (float*)d_out;
    int*   out_idx = (int*)(out_w + Ntok * (size_t)E);

    dim3 block(256);
    dim3 grid((unsigned)(Ntok / TOK_WG));
    moe_gate_topk_kernel<<<grid, block, 0, stream>>>(x, gw, nwt, noise, out_w, out_idx, D);
}